// CARAFE_72576357367966
// MI455X (gfx1250) — compile-verified
//
#include <hip/hip_runtime.h>
#include <hip/hip_bf16.h>

#define BB   4
#define CC   256
#define HH   64
#define WW   64
#define HWs  (HH*WW)      // 4096
#define HID  64
#define KCH  100          // 2*2*5*5
#define NPIX (BB*HWs)     // 16384

typedef float v2f __attribute__((ext_vector_type(2)));
typedef float v8f __attribute__((ext_vector_type(8)));

// ---------------------------------------------------------------------------
// Kernel 1: h_pre[b][o][n] = sum_c w1[o][c] * x[b][c][n]   (64x256 @ 256x4096)
// One wave per 16x16 output tile, V_WMMA_F32_16X16X4_F32, K stepped by 4.
// ---------------------------------------------------------------------------
__global__ __launch_bounds__(128) void gemm1_wmma(const float* __restrict__ x,
                                                  const float* __restrict__ w1,
                                                  float* __restrict__ hpre) {
    const int wave = blockIdx.x * 4 + (threadIdx.x >> 5);
    const int lane = threadIdx.x & 31;
    const int half = lane >> 4;        // 0: lanes 0-15, 1: lanes 16-31
    const int lm   = lane & 15;
    const int nTile =  wave        & 255;   // 256 N-tiles per batch
    const int mTile = (wave >> 8)  & 3;     // 4 M-tiles (64 channels)
    const int b     =  wave >> 10;          // 4 batches
    const int m0 = mTile * 16;
    const int n0 = nTile * 16;
    const float* xb = x + (size_t)b * CC * HWs;

    v8f acc = {};
    for (int k = 0; k < CC; k += 4) {
        const int ka = k + 2 * half;
        v2f a, bm;
        // A: 16x4 f32 tile of w1 (lanes 0-15: K0/K1, lanes 16-31: K2/K3)
        a.x = w1[(m0 + lm) * CC + ka];
        a.y = w1[(m0 + lm) * CC + ka + 1];
        // B: 4x16 f32 tile of x (N across lanes, mirrored K halves)
        bm.x = xb[(size_t)ka       * HWs + n0 + lm];
        bm.y = xb[(size_t)(ka + 1) * HWs + n0 + lm];
        acc = __builtin_amdgcn_wmma_f32_16x16x4_f32(false, a, false, bm,
                                                    (short)0, acc, false, false);
    }
    // D layout: VGPR r -> M = m0 + r + 8*half, N = n0 + lm
    float* hb = hpre + (size_t)b * HID * HWs;
#pragma unroll
    for (int r = 0; r < 8; ++r)
        hb[(size_t)(m0 + r + 8 * half) * HWs + n0 + lm] = acc[r];
}

// ---------------------------------------------------------------------------
// Kernel 2: batch-norm statistics (training mode) -> fused scale/shift
// One block per hidden channel; reduce sum/sumsq over B*H*W = 16384 values.
// ---------------------------------------------------------------------------
__global__ __launch_bounds__(256) void bn_stats(const float* __restrict__ hpre,
                                                const float* __restrict__ gamma,
                                                const float* __restrict__ beta,
                                                float* __restrict__ stats) {
    const int o = blockIdx.x;          // 0..63
    __shared__ float ssum[256], ssq[256];
    float s = 0.f, q = 0.f;
    for (int i = threadIdx.x; i < NPIX; i += 256) {
        const int b = i >> 12;         // /4096
        const int n = i & (HWs - 1);
        const float v = hpre[(size_t)(b * HID + o) * HWs + n];
        s += v; q += v * v;
    }
    ssum[threadIdx.x] = s; ssq[threadIdx.x] = q;
    __syncthreads();
    for (int st = 128; st > 0; st >>= 1) {
        if (threadIdx.x < st) {
            ssum[threadIdx.x] += ssum[threadIdx.x + st];
            ssq [threadIdx.x] += ssq [threadIdx.x + st];
        }
        __syncthreads();
    }
    if (threadIdx.x == 0) {
        const float inv_n = 1.0f / (float)NPIX;
        const float mean  = ssum[0] * inv_n;
        const float var   = ssq[0] * inv_n - mean * mean;
        const float rstd  = rsqrtf(var + 1e-5f);
        const float sc    = gamma[o] * rstd;
        stats[o]       = sc;                    // scale
        stats[HID + o] = beta[o] - mean * sc;   // shift
    }
}

// ---------------------------------------------------------------------------
// Kernel 3: kern_pre[b][m][n] = sum_k w2[m][k] * relu(bn(h_pre[b][k][n])) + b2[m]
// M = 100 padded to 7 tiles of 16; BN+ReLU fused into the B-operand load.
// ---------------------------------------------------------------------------
__global__ __launch_bounds__(128) void gemm2_wmma(const float* __restrict__ hpre,
                                                  const float* __restrict__ stats,
                                                  const float* __restrict__ w2,
                                                  const float* __restrict__ b2,
                                                  float* __restrict__ kern) {
    const int wave = blockIdx.x * 4 + (threadIdx.x >> 5);
    const int lane = threadIdx.x & 31;
    const int half = lane >> 4;
    const int lm   = lane & 15;
    const int nTile = wave & 255;
    const int rem   = wave >> 8;
    const int mTile = rem % 7;         // 7 tiles cover M = 100 (pad to 112)
    const int b     = rem / 7;
    const int m0 = mTile * 16;
    const int n0 = nTile * 16;
    const float* hb = hpre + (size_t)b * HID * HWs;

    v8f acc = {};
    for (int k = 0; k < HID; k += 4) {
        const int ka = k + 2 * half;
        const int ma = m0 + lm;
        v2f a, bm;
        const bool ok = (ma < KCH);
        a.x = ok ? w2[ma * HID + ka]     : 0.f;
        a.y = ok ? w2[ma * HID + ka + 1] : 0.f;
        float v0 = hb[(size_t)ka       * HWs + n0 + lm];
        float v1 = hb[(size_t)(ka + 1) * HWs + n0 + lm];
        bm.x = fmaxf(fmaf(v0, stats[ka],     stats[HID + ka]),     0.f);
        bm.y = fmaxf(fmaf(v1, stats[ka + 1], stats[HID + ka + 1]), 0.f);
        acc = __builtin_amdgcn_wmma_f32_16x16x4_f32(false, a, false, bm,
                                                    (short)0, acc, false, false);
    }
#pragma unroll
    for (int r = 0; r < 8; ++r) {
        const int m = m0 + r + 8 * half;
        if (m < KCH)
            kern[((size_t)b * KCH + m) * HWs + n0 + lm] = acc[r] + b2[m];
    }
}

// ---------------------------------------------------------------------------
// Kernel 4: in-place softmax over each 25-value neighborhood group (4 groups).
// One thread per pixel; loads are coalesced across threads (stride-HW rows).
// ---------------------------------------------------------------------------
__global__ __launch_bounds__(256) void softmax25(float* __restrict__ kern) {
    const int p = blockIdx.x * 256 + threadIdx.x;   // 0..16383
    const int b = p >> 12;
    const int n = p & (HWs - 1);
    float* base = kern + (size_t)b * KCH * HWs + n;
    for (int s = 0; s < 4; ++s) {
        float v[25];
        float mx = -3.0e38f;
#pragma unroll
        for (int q = 0; q < 25; ++q) {
            v[q] = base[(size_t)(s * 25 + q) * HWs];
            mx = fmaxf(mx, v[q]);
        }
        float sum = 0.f;
#pragma unroll
        for (int q = 0; q < 25; ++q) { v[q] = __expf(v[q] - mx); sum += v[q]; }
        const float inv = 1.0f / sum;
#pragma unroll
        for (int q = 0; q < 25; ++q) base[(size_t)(s * 25 + q) * HWs] = v[q] * inv;
    }
}

// ---------------------------------------------------------------------------
// Kernel 5: reassembly + pixel shuffle.
// Block = (h, c, b), 64 threads over w. 5x68 halo of x staged in LDS;
// kern reads hit L2 (6.5 MB resident); each block writes two full
// contiguous 128-float output rows (coalesced pixel shuffle).
// ---------------------------------------------------------------------------
__global__ __launch_bounds__(64) void reassemble(const float* __restrict__ x,
                                                 const float* __restrict__ kern,
                                                 float* __restrict__ out) {
    const int h = blockIdx.x, c = blockIdx.y, b = blockIdx.z;
    const int w = threadIdx.x;
    __shared__ float lx[5][72];        // 5 rows x (64+4) cols, padded
    const float* xc = x + (size_t)(b * CC + c) * HWs;
    for (int idx = threadIdx.x; idx < 5 * 68; idx += 64) {
        const int r  = idx / 68;
        const int cc = idx % 68;
        const int gr = h + r - 2;
        const int gc = cc - 2;
        float v = 0.f;
        if (gr >= 0 && gr < HH && gc >= 0 && gc < WW) v = xc[gr * WW + gc];
        lx[r][cc] = v;
    }
    __syncthreads();

    const float* kb = kern + (size_t)b * KCH * HWs + h * WW + w;
    float acc0 = 0.f, acc1 = 0.f, acc2 = 0.f, acc3 = 0.f;
#pragma unroll
    for (int q = 0; q < 25; ++q) {
        const int di = q / 5, dj = q % 5;
        const float xv = lx[di][w + dj];
        acc0 = fmaf(kb[(size_t)( 0 + q) * HWs], xv, acc0);   // s = (0,0)
        acc1 = fmaf(kb[(size_t)(25 + q) * HWs], xv, acc1);   // s = (0,1)
        acc2 = fmaf(kb[(size_t)(50 + q) * HWs], xv, acc2);   // s = (1,0)
        acc3 = fmaf(kb[(size_t)(75 + q) * HWs], xv, acc3);   // s = (1,1)
    }
    float* ob = out + (size_t)(b * CC + c) * (2 * HH) * (2 * WW);
    ob[(2 * h + 0) * (2 * WW) + 2 * w + 0] = acc0;
    ob[(2 * h + 0) * (2 * WW) + 2 * w + 1] = acc1;
    ob[(2 * h + 1) * (2 * WW) + 2 * w + 0] = acc2;
    ob[(2 * h + 1) * (2 * WW) + 2 * w + 1] = acc3;
}

// ---------------------------------------------------------------------------
extern "C" void kernel_launch(void* const* d_in, const int* in_sizes, int n_in,
                              void* d_out, int out_size, void* d_ws, size_t ws_size,
                              hipStream_t stream) {
    (void)in_sizes; (void)n_in; (void)out_size; (void)ws_size;
    const float* x     = (const float*)d_in[0];
    const float* w1    = (const float*)d_in[1];
    const float* gamma = (const float*)d_in[2];
    const float* beta  = (const float*)d_in[3];
    const float* w2    = (const float*)d_in[4];
    const float* b2    = (const float*)d_in[5];
    float* out = (float*)d_out;

    float* ws    = (float*)d_ws;
    float* hpre  = ws;                                   // B*HID*HW  = 1,048,576 f
    float* stats = ws + (size_t)BB * HID * HWs;          // 2*HID     = 128 f
    float* kern  = stats + 2 * HID;                      // B*KCH*HW  = 1,638,400 f

    // 1) encoder GEMM1: 4096 waves (B * 4 Mtiles * 256 Ntiles), 4 waves/block
    gemm1_wmma<<<1024, 128, 0, stream>>>(x, w1, hpre);
    // 2) BN batch statistics -> fused scale/shift
    bn_stats<<<HID, 256, 0, stream>>>(hpre, gamma, beta, stats);
    // 3) encoder GEMM2 (+BN+ReLU fused in, +bias): 7168 waves
    gemm2_wmma<<<1792, 128, 0, stream>>>(hpre, stats, w2, b2, kern);
    // 4) softmax over 25-neighborhoods, in place
    softmax25<<<NPIX / 256, 256, 0, stream>>>(kern);
    // 5) content-aware reassembly + pixel shuffle
    reassemble<<<dim3(HH, CC, BB), 64, 0, stream>>>(x, kern, out);
}